// NonFirstLayerAggregator_35966056136851
// MI455X (gfx1250) — compile-verified
//
#include <hip/hip_runtime.h>

// Problem constants from the reference
#define KNEIGH 10
#define DIM    64
#define FEATS  192   // 3 * DIM

// Native clang vector type: supports elementwise ops and is accepted by
// __builtin_nontemporal_load/store (HIP's float4 class is not).
typedef __attribute__((ext_vector_type(4))) float v4f;

__global__ __launch_bounds__(256) void
nonfirst_layer_agg_kernel(const int* __restrict__ nodes,
                          const int* __restrict__ neigh_pos,
                          const int* __restrict__ neigh_neg,
                          const int* __restrict__ len_pos,
                          const int* __restrict__ len_neg,
                          const float* __restrict__ E_bal,
                          const float* __restrict__ E_unbal,
                          float* __restrict__ out,
                          int B)
{
    // wave32: two nodes per wave, half-wave (16 lanes) per node,
    // each lane owns 4 of the 64 embedding dims (v4f / B128 accesses).
    const int lane  = threadIdx.x & 31;
    const int wave  = (blockIdx.x * blockDim.x + threadIdx.x) >> 5;
    const int half  = lane >> 4;           // which node of the pair
    const int sub   = lane & 15;           // 16 dim-groups
    const int b     = wave * 2 + half;
    if (b >= B) return;

    const int dbase = sub * 4;             // dims [dbase .. dbase+3]

    const int lp = len_pos[b];
    const int ln = len_neg[b];

    v4f acc_pb = {0.f, 0.f, 0.f, 0.f};  // pos @ E_bal
    v4f acc_pu = {0.f, 0.f, 0.f, 0.f};  // pos @ E_unbal
    v4f acc_nb = {0.f, 0.f, 0.f, 0.f};  // neg @ E_bal
    v4f acc_nu = {0.f, 0.f, 0.f, 0.f};  // neg @ E_unbal

    const int nb_base = b * KNEIGH;

    // Fully unrolled gather loop: up to 40 predicated global_load_b128 per
    // node-pair in flight. EXEC-masked lanes (k >= len) issue no memory
    // requests, so invalid neighbors cost zero HBM bandwidth.
#pragma unroll
    for (int k = 0; k < KNEIGH; ++k) {
        if (k < lp) {
            const int off = neigh_pos[nb_base + k] * DIM + dbase;
            acc_pb += *reinterpret_cast<const v4f*>(E_bal   + off);
            acc_pu += *reinterpret_cast<const v4f*>(E_unbal + off);
        }
        if (k < ln) {
            const int off = neigh_neg[nb_base + k] * DIM + dbase;
            acc_nb += *reinterpret_cast<const v4f*>(E_bal   + off);
            acc_nu += *reinterpret_cast<const v4f*>(E_unbal + off);
        }
    }

    const float rp = 1.0f / (float)(lp > 1 ? lp : 1);
    const float rn = 1.0f / (float)(ln > 1 ? ln : 1);
    acc_pb *= rp; acc_pu *= rp;
    acc_nb *= rn; acc_nu *= rn;

    // Self rows
    const int self_off = nodes[b] * DIM + dbase;
    const v4f self_b = *reinterpret_cast<const v4f*>(E_bal   + self_off);
    const v4f self_u = *reinterpret_cast<const v4f*>(E_unbal + self_off);

    // Outputs (streaming, written once): non-temporal B128 stores so the
    // 154 MB output stream does not evict embedding lines from the 192 MB L2.
    // to_feats_bal   = [pos_bal | neg_unbal | self_bal]
    // to_feats_unbal = [pos_unbal | neg_bal | self_unbal]
    float* ob = out + (size_t)b * FEATS;
    float* ou = out + (size_t)B * FEATS + (size_t)b * FEATS;

    __builtin_nontemporal_store(acc_pb, reinterpret_cast<v4f*>(ob + dbase));
    __builtin_nontemporal_store(acc_nu, reinterpret_cast<v4f*>(ob + DIM + dbase));
    __builtin_nontemporal_store(self_b, reinterpret_cast<v4f*>(ob + 2 * DIM + dbase));

    __builtin_nontemporal_store(acc_pu, reinterpret_cast<v4f*>(ou + dbase));
    __builtin_nontemporal_store(acc_nb, reinterpret_cast<v4f*>(ou + DIM + dbase));
    __builtin_nontemporal_store(self_u, reinterpret_cast<v4f*>(ou + 2 * DIM + dbase));
}

extern "C" void kernel_launch(void* const* d_in, const int* in_sizes, int n_in,
                              void* d_out, int out_size, void* d_ws, size_t ws_size,
                              hipStream_t stream) {
    const int*   nodes     = (const int*)  d_in[0];
    const int*   neigh_pos = (const int*)  d_in[1];
    const int*   neigh_neg = (const int*)  d_in[2];
    const int*   len_pos   = (const int*)  d_in[3];
    const int*   len_neg   = (const int*)  d_in[4];
    const float* E_bal     = (const float*)d_in[5];
    const float* E_unbal   = (const float*)d_in[6];
    float*       out       = (float*)      d_out;

    const int B = in_sizes[0];            // number of nodes (100000)

    // 2 nodes per wave32 -> 16 nodes per 256-thread block
    const int threads = 256;
    const int nodes_per_block = (threads / 32) * 2;
    const int blocks = (B + nodes_per_block - 1) / nodes_per_block;

    nonfirst_layer_agg_kernel<<<blocks, threads, 0, stream>>>(
        nodes, neigh_pos, neigh_neg, len_pos, len_neg, E_bal, E_unbal, out, B);
}